// PREF_18863496364435
// MI455X (gfx1250) — compile-verified
//
#include <hip/hip_runtime.h>
#include <math.h>

typedef __attribute__((ext_vector_type(2))) float v2f;
typedef __attribute__((ext_vector_type(8))) float v8f;

// Explicit global-address-space pointers: keep VMEM on the GLOBAL path
// (LOADcnt/STOREcnt only), never FLAT (which also ties up DScnt / LDS path).
typedef const __attribute__((address_space(1))) float* cgfptr;
typedef       __attribute__((address_space(1))) float* gfptr;
typedef const __attribute__((address_space(1))) v2f*   cgv2fptr;
typedef       __attribute__((address_space(1))) v2f*   gv2fptr;

__device__ __forceinline__ cgfptr as_global(const float* p) {
    return (cgfptr)(unsigned long long)p;
}
__device__ __forceinline__ gfptr as_global_mut(float* p) {
    return (gfptr)(unsigned long long)p;
}

#define NSPAT 256
#define NCH   128      // DF(16) * D(8)
#define NPTS_TOTAL 131072
#define TWO_PI 6.28318530717958647692f

__device__ __forceinline__ float softplus_b10(float a) {
    float bx = 10.0f * a;
    if (bx > 1.0f) return a;
    float e = expf(fminf(bx, 30.0f));
    return 0.1f * log1pf(e);
}

// ---------------------------------------------------------------------------
// Build DFT matrix W[m][k] = exp(+2*pi*i*m*k/256) (ifft convention, no scale)
// plus an alpha-pre-scaled copy (alpha = softplus_b10(alpha_params[0])), so
// pass-1 GEMMs need no per-fragment scaling at all.
// Uses (m*k) mod 256 so the trig argument stays in [0, 2*pi) exactly.
// ---------------------------------------------------------------------------
__global__ void build_w(float* __restrict__ Wr,  float* __restrict__ Wi,
                        float* __restrict__ Wra, float* __restrict__ Wia,
                        const float* __restrict__ alpha_param) {
    int idx = blockIdx.x * blockDim.x + threadIdx.x;   // 65536 total
    int m = idx >> 8;
    int k = idx & 255;
    int p = (m * k) & 255;
    float ang = (TWO_PI / 256.0f) * (float)p;
    float cr = cosf(ang);
    float si = sinf(ang);
    float alpha = softplus_b10(alpha_param[0]);
    Wr[idx]  = cr;
    Wi[idx]  = si;
    Wra[idx] = cr * alpha;
    Wia[idx] = si * alpha;
}

// ---------------------------------------------------------------------------
// Batched complex GEMM:  D_c[m,n] = sum_k A[m,k] * X_c[n,k]   (m,n,k in 0..255)
// A = (Ar,Ai) is the (optionally alpha-scaled) DFT matrix, row-major 256x256.
// X_c element (n,k) at:  X + chanBase(c) + n*rowStride + k*KSTR
//   chanBase(c) = (c>>3)*fStride + (c&7)*kSmall
// Output:
//   outS == nullptr : planar  outR/outI[c*65536 + m*256 + n]
//   outS != nullptr : interleaved outS[((m*256+n)*128 + c)*2 + {0,1}]
// One wave computes a 16(M)x64(N) complex tile. K-loop: 16 chunks x 4 steps,
// all load offsets are compile-time immediates; per step the 10 loads are
// issued ahead of the 16 V_WMMA_F32_16X16X4_F32 so loads clause and overlap.
// ---------------------------------------------------------------------------
template<int KSTR>
__global__ __launch_bounds__(256)
void dft_gemm(const float* __restrict__ Ar, const float* __restrict__ Ai,
              const float* __restrict__ Xr, const float* __restrict__ Xi,
              long fStride, long kSmall, long rowStride,
              float* __restrict__ outR, float* __restrict__ outI,
              float* __restrict__ outS)
{
    int bx     = blockIdx.x;
    int nGroup = bx & 3;          // 4 groups of 64 columns
    int mGroup = (bx >> 2) & 1;   // 2 groups of 128 rows (8 waves x 16)
    int c      = bx >> 3;         // channel 0..127
    int wave   = threadIdx.x >> 5;
    int lane   = threadIdx.x & 31;
    int m0    = (mGroup * 8 + wave) * 16;
    int nBase = nGroup * 64;

    int lrow  = lane & 15;        // M for A-frag, N for B-frag
    int khalf = (lane >> 4) * 2;  // K base within group of 4

    // A streams: W row (m0+lrow), khalf folded into the base (8B aligned)
    cgfptr awr = as_global(Ar + (long)(m0 + lrow) * 256 + khalf);
    cgfptr awi = as_global(Ai + (long)(m0 + lrow) * 256 + khalf);

    long cb = (long)(c >> 3) * fStride + (long)(c & 7) * kSmall;
    cgfptr bxr[4]; cgfptr bxi[4];
    #pragma unroll
    for (int j = 0; j < 4; ++j) {
        long ro = cb + (long)(nBase + j * 16 + lrow) * rowStride
                     + (long)khalf * KSTR;
        __builtin_prefetch(Xr + ro, 0, 3);   // global_prefetch_b8
        __builtin_prefetch(Xi + ro, 0, 3);
        bxr[j] = as_global(Xr + ro);
        bxi[j] = as_global(Xi + ro);
    }

    v8f accR[4], accI[4];
    #pragma unroll
    for (int j = 0; j < 4; ++j) {
        accR[j] = (v8f){0.f,0.f,0.f,0.f,0.f,0.f,0.f,0.f};
        accI[j] = (v8f){0.f,0.f,0.f,0.f,0.f,0.f,0.f,0.f};
    }

    #pragma unroll 1
    for (int kc = 0; kc < 16; ++kc) {        // 16 chunks x (4 steps x K=4)
        #pragma unroll
        for (int s = 0; s < 4; ++s) {
            // ---- all loads for this step (constant immediate offsets) ----
            v2f a_r = *(cgv2fptr)(awr + s * 8);
            v2f a_i = *(cgv2fptr)(awi + s * 8);
            v2f br[4], bi[4];
            #pragma unroll
            for (int j = 0; j < 4; ++j) {
                if (KSTR == 1) {
                    br[j] = *(cgv2fptr)(bxr[j] + s * 4);
                    bi[j] = *(cgv2fptr)(bxi[j] + s * 4);
                } else {
                    br[j] = (v2f){ bxr[j][s * 4 * KSTR], bxr[j][s * 4 * KSTR + KSTR] };
                    bi[j] = (v2f){ bxi[j][s * 4 * KSTR], bxi[j][s * 4 * KSTR + KSTR] };
                }
            }
            v2f a_m = -a_i;   // for -Wi*Xi term (f32 WMMA NEG only covers C)
            // ---- 16 WMMAs ----
            #pragma unroll
            for (int j = 0; j < 4; ++j) {
                accR[j] = __builtin_amdgcn_wmma_f32_16x16x4_f32(false, a_r, false, br[j],
                                                                (short)0, accR[j], false, false);
                accR[j] = __builtin_amdgcn_wmma_f32_16x16x4_f32(false, a_m, false, bi[j],
                                                                (short)0, accR[j], false, false);
                accI[j] = __builtin_amdgcn_wmma_f32_16x16x4_f32(false, a_i, false, br[j],
                                                                (short)0, accI[j], false, false);
                accI[j] = __builtin_amdgcn_wmma_f32_16x16x4_f32(false, a_r, false, bi[j],
                                                                (short)0, accI[j], false, false);
            }
        }
        awr += 32; awi += 32;                // 4 steps * 8 floats
        #pragma unroll
        for (int j = 0; j < 4; ++j) {
            bxr[j] += 16 * KSTR;
            bxi[j] += 16 * KSTR;
        }
    }

    // C/D layout: VGPR r, lanes 0-15 -> M=r, lanes 16-31 -> M=r+8; N = lane&15
    int mBase = m0 + ((lane >> 4) << 3);
    if (outS) {
        gfptr outSg = as_global_mut(outS);
        #pragma unroll
        for (int j = 0; j < 4; ++j) {
            int nIdx = nBase + j * 16 + lrow;
            #pragma unroll
            for (int r = 0; r < 8; ++r) {
                long m = mBase + r;
                v2f val = { accR[j][r], accI[j][r] };
                *(gv2fptr)(outSg + ((m * 256 + nIdx) * NCH + c) * 2) = val;
            }
        }
    } else {
        gfptr outRg = as_global_mut(outR);
        gfptr outIg = as_global_mut(outI);
        long cOff = (long)c * 65536;
        #pragma unroll
        for (int j = 0; j < 4; ++j) {
            int nIdx = nBase + j * 16 + lrow;
            #pragma unroll
            for (int r = 0; r < 8; ++r) {
                long m = mBase + r;
                outRg[cOff + m * 256 + nIdx] = accR[j][r];
                outIg[cOff + m * 256 + nIdx] = accI[j][r];
            }
        }
    }
}

// ---------------------------------------------------------------------------
// Per-point bilinear sample of 3 interleaved fields + small irfft + sum.
// One wave per point; lanes cover 32 channels per chunk (4 chunks of 32).
// Channel c = d*8 + k  (d = feature, k = frequency).
// ---------------------------------------------------------------------------
__device__ __forceinline__ void bilin_setup(float gh, float gw,
                                            long* off, float* w)
{
    float ih = (gh + 1.0f) * 0.5f * 255.0f;
    float iw = (gw + 1.0f) * 0.5f * 255.0f;
    float h0f = floorf(ih), w0f = floorf(iw);
    float wh = ih - h0f, ww = iw - w0f;
    int h0 = (int)h0f; h0 = h0 < 0 ? 0 : (h0 > 255 ? 255 : h0);
    int w0 = (int)w0f; w0 = w0 < 0 ? 0 : (w0 > 255 ? 255 : w0);
    int h1 = h0 + 1 > 255 ? 255 : h0 + 1;
    int w1 = w0 + 1 > 255 ? 255 : w0 + 1;
    // *NCH complex *2 floats = *256
    off[0] = ((long)h0 * 256 + w0) * 256;
    off[1] = ((long)h0 * 256 + w1) * 256;
    off[2] = ((long)h1 * 256 + w0) * 256;
    off[3] = ((long)h1 * 256 + w1) * 256;
    w[0] = (1.0f - wh) * (1.0f - ww);
    w[1] = (1.0f - wh) * ww;
    w[2] = wh * (1.0f - ww);
    w[3] = wh * ww;
}

__global__ __launch_bounds__(256)
void sample_kernel(const float* __restrict__ pts,
                   const float* __restrict__ Sx,
                   const float* __restrict__ Sy,
                   const float* __restrict__ Sz,
                   float* __restrict__ out)
{
    int wavesPerBlock = blockDim.x >> 5;
    int gwave = blockIdx.x * wavesPerBlock + (threadIdx.x >> 5);
    int lane  = threadIdx.x & 31;
    if (gwave >= NPTS_TOTAL) return;

    cgfptr ptsg = as_global(pts);
    float x = ptsg[gwave * 3 + 0];
    float y = ptsg[gwave * 3 + 1];
    float z = ptsg[gwave * 3 + 2];

    // Twiddles for this lane's frequency k = lane&7, per axis (x,y,z order)
    int k = lane & 7;
    float coords[3] = { x, y, z };
    float twr[3], twi[3];
    #pragma unroll
    for (int a = 0; a < 3; ++a) {
        float xx  = (coords[a] + 1.0f) * 0.5f * (255.0f / 256.0f);
        float ang = TWO_PI * xx * (float)k;
        float s   = (k > 0) ? 2.0f : 1.0f;
        twr[a] = cosf(ang) * s;
        twi[a] = sinf(ang) * s;
    }

    // Bilinear footprints: Sx[y][z][c], Sy[x][z][c], Sz[x][y][c]
    long off[3][4]; float wgt[3][4];
    bilin_setup(y, z, off[0], wgt[0]);
    bilin_setup(x, z, off[1], wgt[1]);
    bilin_setup(x, y, off[2], wgt[2]);
    cgfptr fld[3] = { as_global(Sx), as_global(Sy), as_global(Sz) };

    gfptr outg = as_global_mut(out);

    #pragma unroll
    for (int chunk = 0; chunk < 4; ++chunk) {
        int c = chunk * 32 + lane;
        long co = (long)c * 2;
        float acc = 0.0f;
        #pragma unroll
        for (int a = 0; a < 3; ++a) {
            cgfptr s = fld[a];
            v2f v00 = *(cgv2fptr)(s + off[a][0] + co);
            v2f v01 = *(cgv2fptr)(s + off[a][1] + co);
            v2f v10 = *(cgv2fptr)(s + off[a][2] + co);
            v2f v11 = *(cgv2fptr)(s + off[a][3] + co);
            float vre = v00.x*wgt[a][0] + v01.x*wgt[a][1] + v10.x*wgt[a][2] + v11.x*wgt[a][3];
            float vim = v00.y*wgt[a][0] + v01.y*wgt[a][1] + v10.y*wgt[a][2] + v11.y*wgt[a][3];
            acc += vre * twr[a] - vim * twi[a];
        }
        // reduce over the 8 frequency lanes (k = lane&7)
        acc += __shfl_xor(acc, 1, 32);
        acc += __shfl_xor(acc, 2, 32);
        acc += __shfl_xor(acc, 4, 32);
        if ((lane & 7) == 0) {
            int d = chunk * 4 + (lane >> 3);
            outg[(long)gwave * 16 + d] = acc;
        }
    }
}

// ---------------------------------------------------------------------------
extern "C" void kernel_launch(void* const* d_in, const int* in_sizes, int n_in,
                              void* d_out, int out_size, void* d_ws, size_t ws_size,
                              hipStream_t stream)
{
    (void)in_sizes; (void)n_in; (void)out_size; (void)ws_size;
    const float* pts   = (const float*)d_in[0];
    const float* Fx_re = (const float*)d_in[1];
    const float* Fx_im = (const float*)d_in[2];
    const float* Fy_re = (const float*)d_in[3];
    const float* Fy_im = (const float*)d_in[4];
    const float* Fz_re = (const float*)d_in[5];
    const float* Fz_im = (const float*)d_in[6];
    const float* alpha = (const float*)d_in[7];

    float* ws  = (float*)d_ws;
    float* Wr  = ws;                     // 65536 floats
    float* Wi  = Wr  + 65536;            // 65536
    float* Wra = Wi  + 65536;            // 65536 (alpha-scaled)
    float* Wia = Wra + 65536;            // 65536
    float* Ur  = Wia + 65536;            // 128*65536 = 8388608
    float* Ui  = Ur  + 8388608;          // 8388608
    float* Sx  = Ui  + 8388608;          // 16777216 (interleaved complex)
    float* Sy  = Sx  + 16777216;
    float* Sz  = Sy  + 16777216;

    build_w<<<256, 256, 0, stream>>>(Wr, Wi, Wra, Wia, alpha);

    const int gemmGrid = 128 * 4 * 2;    // channels * nGroups * mGroups

    // Field X: Fx[f][k1][ky][kz]; plane P[a=ky][b=kz]
    dft_gemm<1><<<gemmGrid, 256, 0, stream>>>(Wra, Wia, Fx_re, Fx_im,
        524288, 65536, 256, Ur, Ui, nullptr);
    dft_gemm<1><<<gemmGrid, 256, 0, stream>>>(Wr, Wi, Ur, Ui,
        524288, 65536, 256, nullptr, nullptr, Sx);

    // Field Y: Fy[f][kx][k2][kz]; plane P[a=kx][b=kz]
    dft_gemm<1><<<gemmGrid, 256, 0, stream>>>(Wra, Wia, Fy_re, Fy_im,
        524288, 256, 2048, Ur, Ui, nullptr);
    dft_gemm<1><<<gemmGrid, 256, 0, stream>>>(Wr, Wi, Ur, Ui,
        524288, 65536, 256, nullptr, nullptr, Sy);

    // Field Z: Fz[f][kx][ky][k3]; plane P[a=kx][b=ky], b-stride 8
    dft_gemm<8><<<gemmGrid, 256, 0, stream>>>(Wra, Wia, Fz_re, Fz_im,
        524288, 1, 2048, Ur, Ui, nullptr);
    dft_gemm<1><<<gemmGrid, 256, 0, stream>>>(Wr, Wi, Ur, Ui,
        524288, 65536, 256, nullptr, nullptr, Sz);

    // Sampling + irfft: one wave per point, 8 waves per block
    sample_kernel<<<NPTS_TOTAL / 8, 256, 0, stream>>>(pts, Sx, Sy, Sz, (float*)d_out);
}